// WordGraphAttention_51075751084517
// MI455X (gfx1250) — compile-verified
//
#include <hip/hip_runtime.h>
#include <hip/hip_bf16.h>
#include <math.h>

// Shapes: B=32, S=128, QD=768, N1=16, M1=32, M2=32, D=100
#define Bv 32
#define Sv 128
#define QDv 768
#define N1v 16
#define M1v 32
#define M2v 32
#define Dv 100

typedef float v2f __attribute__((ext_vector_type(2)));
typedef float v4f __attribute__((ext_vector_type(4)));
typedef float v8f __attribute__((ext_vector_type(8)));

__device__ __forceinline__ float wave_sum(float v) {
#pragma unroll
  for (int off = 16; off > 0; off >>= 1) v += __shfl_xor(v, off, 32);
  return v;
}
__device__ __forceinline__ float wave_max(float v) {
#pragma unroll
  for (int off = 16; off > 0; off >>= 1) v = fmaxf(v, __shfl_xor(v, off, 32));
  return v;
}

// ---------------------------------------------------------------------------
// Kernel 1: Qt_p = tanh(Q @ W_q_p^T + b_q_p), Q = q[:,0,:] (32 x 768)
// WMMA f32 16x16x4. 28 blocks of 1 wave: p in {0,1}, 2 M-tiles x 7 N-tiles.
// Padded N columns (100..111) load a clamped-but-valid row; their garbage
// lands only in C columns we never store. No branches in the K loop.
// ---------------------------------------------------------------------------
__global__ __launch_bounds__(32) void proj_tanh_wmma(
    const float* __restrict__ q,
    const float* __restrict__ W_q2, const float* __restrict__ b_q2,
    const float* __restrict__ W_q1, const float* __restrict__ b_q1,
    float* __restrict__ qt2, float* __restrict__ qt1) {
  int bid = blockIdx.x;           // 0..27
  int p   = bid / 14;             // 0 -> proj2, 1 -> proj1
  int t   = bid % 14;
  int mt  = t / 7;                // M tile (rows of Q, i.e. batch)
  int nt  = t % 7;                // N tile (output feature)
  const float* W    = p ? W_q1 : W_q2;
  const float* bias = p ? b_q1 : b_q2;
  float*       Qt   = p ? qt1  : qt2;

  int lane = threadIdx.x;
  int Mloc = lane & 15;
  int kh   = (lane >> 4) << 1;    // lanes 16-31 hold K+2,K+3
  int qrowIdx = mt * 16 + Mloc;   // batch row 0..31
  int col     = nt * 16 + Mloc;   // B-operand column (output feature)
  int colC    = (col < Dv) ? col : 0;   // clamped, always in-bounds
  const float* qrow = q + (size_t)qrowIdx * (Sv * QDv) + kh;  // s = 0 slice
  const float* wrow = W + (size_t)colC   * QDv        + kh;

  v8f c = {};
#pragma unroll 4
  for (int kb = 0; kb < QDv / 4; ++kb) {
    v2f a, b;
    a.x = qrow[0]; a.y = qrow[1];
    b.x = wrow[0]; b.y = wrow[1];
    qrow += 4; wrow += 4;
    c = __builtin_amdgcn_wmma_f32_16x16x4_f32(false, a, false, b,
                                              (short)0, c, false, false);
  }

  // C layout: VGPR r -> lanes 0-15: M=r, N=lane; lanes 16-31: M=r+8, N=lane-16
  int nOut  = nt * 16 + (lane & 15);
  int rbase = mt * 16 + ((lane >> 4) << 3);
  if (nOut < Dv) {
    float bv = bias[nOut];
#pragma unroll
    for (int r = 0; r < 8; ++r)
      Qt[(size_t)(rbase + r) * Dv + nOut] = tanhf(c[r] + bv);
  }
}

// ---------------------------------------------------------------------------
// Kernel 2: QW_p = Qt_p @ W_kv_p   (32x100 @ 100x100), WMMA f32 16x16x4
// ---------------------------------------------------------------------------
__global__ __launch_bounds__(32) void qw_wmma(
    const float* __restrict__ qt2, const float* __restrict__ qt1,
    const float* __restrict__ W_kv2, const float* __restrict__ W_kv1,
    float* __restrict__ qw2, float* __restrict__ qw1) {
  int bid = blockIdx.x;           // 0..27
  int p   = bid / 14;
  int t   = bid % 14;
  int mt  = t / 7;
  int nt  = t % 7;
  const float* A = p ? qt1 : qt2;       // 32 x 100 row-major (K contiguous)
  const float* W = p ? W_kv1 : W_kv2;   // B[e][d] = W[e*100 + d]
  float*       O = p ? qw1 : qw2;

  int lane = threadIdx.x;
  int Mloc = lane & 15;
  int kh   = (lane >> 4) << 1;
  int row  = mt * 16 + Mloc;
  int col  = nt * 16 + Mloc;
  int colC = (col < Dv) ? col : 0;      // clamped column; garbage discarded
  const float* arow = A + (size_t)row * Dv + kh;

  v8f c = {};
#pragma unroll
  for (int kb = 0; kb < Dv / 4; ++kb) {   // 25 exact K-steps
    int k0 = kb * 4 + kh;
    v2f a, b;
    a.x = arow[0]; a.y = arow[1];
    arow += 4;
    b.x = W[(size_t)k0 * Dv + colC];
    b.y = W[(size_t)(k0 + 1) * Dv + colC];
    c = __builtin_amdgcn_wmma_f32_16x16x4_f32(false, a, false, b,
                                              (short)0, c, false, false);
  }

  int nOut  = nt * 16 + (lane & 15);
  int rbase = mt * 16 + ((lane >> 4) << 3);
  if (nOut < Dv) {
#pragma unroll
    for (int r = 0; r < 8; ++r)
      O[(size_t)(rbase + r) * Dv + nOut] = c[r];
  }
}

// ---------------------------------------------------------------------------
// Kernel 3: fused attention over one (b, n1) tile per block (256 thr = 8 waves)
// Streams k_2/v_2/k_1/v_1 once (~420 MB total -> HBM-bound, ~18 us floor).
// Each lane owns a contiguous 4-float slice of the D=100 rows so every row
// is consumed by one global_load_b128 per lane (lanes 25..31 neutralized by
// zero-padded LDS weights + clamped in-bounds addresses; no exec branches).
// ---------------------------------------------------------------------------
__global__ __launch_bounds__(256) void wordgraph_attn_main(
    const float* __restrict__ k_1, const float* __restrict__ v_1,
    const float* __restrict__ k_2, const float* __restrict__ v_2,
    const float* __restrict__ qw2, const float* __restrict__ qw1,
    float* __restrict__ out) {
  const float inv_sqrt_d = 0.1f;      // 1/sqrt(100)
  const int bn = blockIdx.x;          // 0..511
  const int b  = bn / N1v;

  const float* k2 = k_2 + (size_t)bn * (M1v * M2v * Dv);
  const float* v2 = v_2 + (size_t)bn * (M1v * M2v * Dv);
  const float* k1 = k_1 + (size_t)bn * (M1v * Dv);
  const float* v1 = v_1 + (size_t)bn * (M1v * Dv);

  __shared__ __align__(16) float s_qw2[128];   // zero-padded beyond D
  __shared__ __align__(16) float s_qw1[128];
  __shared__ float s_att[M1v * M2v];           // 1024
  __shared__ __align__(16) float s_reps[M1v][Dv];  // masked reps2 (row=400B)
  __shared__ float s_att1[M1v];
  __shared__ float s_vmask[M1v];

  const int tid  = threadIdx.x;
  const int wave = tid >> 5;
  const int lane = tid & 31;

  if (tid < 128) {
    s_qw2[tid] = (tid < Dv) ? qw2[(size_t)b * Dv + tid] : 0.0f;
    s_qw1[tid] = (tid < Dv) ? qw1[(size_t)b * Dv + tid] : 0.0f;
  }
  __syncthreads();

  const bool laneOk = (lane < 25);             // 25 lanes * 4 = 100 = D
  const int  d4     = laneOk ? (lane * 4) : 0; // clamped, 16B aligned

  // ---- Phase 1: att2[m,k] = <k_2 row, qw2> * inv_sqrt_d (1024 rows/block)
  {
    const v4f qv = *(const v4f*)&s_qw2[lane * 4];  // zeros for lane >= 25
    for (int r = wave; r < M1v * M2v; r += 8) {
      const float* row = k2 + (size_t)r * Dv;
      int rn = (r + 8 < M1v * M2v) ? (r + 8) : r;
      __builtin_prefetch(k2 + (size_t)rn * Dv + d4, 0, 1);
      const v4f rv = *(const v4f*)(row + d4);
      float s = rv.x * qv.x + rv.y * qv.y + rv.z * qv.z + rv.w * qv.w;
      s = wave_sum(s);
      if (lane == 0) s_att[r] = s * inv_sqrt_d;
    }
  }
  __syncthreads();

  // ---- Phase 2: masked softmax over k (32 lanes == M2, wave32 exact fit)
  for (int m = wave; m < M1v; m += 8) {
    float a = s_att[m * M2v + lane];
    a = (a == 0.0f) ? -10000.0f : a;
    a = (a > 0.0f) ? a : 0.01f * a;          // leaky_relu(.,0.01)
    float mx = wave_max(a);
    float e  = __expf(a - mx);
    float sm = wave_sum(e);
    float pr = e / sm;
    pr = (pr == 0.03125f) ? 0.0f : pr;       // == 1/32 -> 0
    s_att[m * M2v + lane] = pr;
  }
  __syncthreads();

  // ---- Phase 3: V_mask[m], reps2[m,:] = mask * sum_k p[k] * v_2[m,k,:]
  for (int m = wave; m < M1v; m += 8) {
    const float* v1r = v1 + (size_t)m * Dv;
    const v4f v1v = *(const v4f*)(v1r + d4);
    float vs = laneOk ? (v1v.x + v1v.y + v1v.z + v1v.w) : 0.0f;
    vs = wave_sum(vs);
    float mask = (vs != 0.0f) ? 1.0f : 0.0f;
    if (lane == 0) s_vmask[m] = mask;

    v4f acc = {};
    const float* v2m = v2 + (size_t)m * (M2v * Dv);
#pragma unroll 4
    for (int k = 0; k < M2v; ++k) {
      float p = s_att[m * M2v + k];
      const v4f vv = *(const v4f*)(v2m + (size_t)k * Dv + d4);
      acc += p * vv;                        // one b128 load + packed FMAs
    }
    if (laneOk) *(v4f*)&s_reps[m][lane * 4] = mask * acc;
  }
  __syncthreads();

  // ---- Phase 4: att1[m] = <k_1 row, qw1> * inv_sqrt_d, then softmax over m
  {
    const v4f pv = *(const v4f*)&s_qw1[lane * 4];  // zeros for lane >= 25
    for (int m = wave; m < M1v; m += 8) {
      const v4f kv = *(const v4f*)(k1 + (size_t)m * Dv + d4);
      float s = kv.x * pv.x + kv.y * pv.y + kv.z * pv.z + kv.w * pv.w;
      s = wave_sum(s);
      if (lane == 0) s_att1[m] = s * inv_sqrt_d;
    }
  }
  __syncthreads();

  if (wave == 0) {
    float a = s_att1[lane];
    a = (a == 0.0f) ? -10000.0f : a;
    a = (a > 0.0f) ? a : 0.01f * a;
    float mx = wave_max(a);
    float e  = __expf(a - mx);
    float sm = wave_sum(e);
    float pr = e / sm;
    pr = (pr == 0.03125f) ? 0.0f : pr;
    s_att1[lane] = pr;
  }
  __syncthreads();

  // ---- Phase 5: out[b,n,e] = sum_m att1[m] * V1[m,e], V1 = [mask*v1, reps2]
  if (tid < 2 * Dv) {
    float acc = 0.0f;
    if (tid < Dv) {
#pragma unroll 4
      for (int m = 0; m < M1v; ++m)
        acc += s_att1[m] * s_vmask[m] * v1[(size_t)m * Dv + tid];
    } else {
      int d = tid - Dv;
#pragma unroll 4
      for (int m = 0; m < M1v; ++m)
        acc += s_att1[m] * s_reps[m][d];     // mask already applied
    }
    out[(size_t)bn * (2 * Dv) + tid] = acc;
  }
}

// ---------------------------------------------------------------------------
extern "C" void kernel_launch(void* const* d_in, const int* in_sizes, int n_in,
                              void* d_out, int out_size, void* d_ws, size_t ws_size,
                              hipStream_t stream) {
  const float* q     = (const float*)d_in[0];
  const float* k_1   = (const float*)d_in[1];
  const float* v_1   = (const float*)d_in[2];
  const float* k_2   = (const float*)d_in[3];
  const float* v_2   = (const float*)d_in[4];
  /* d_in[5] = input_ent : unused by the reference */
  const float* W_kv1 = (const float*)d_in[6];
  const float* W_kv2 = (const float*)d_in[7];
  const float* W_q1  = (const float*)d_in[8];
  const float* b_q1  = (const float*)d_in[9];
  const float* W_q2  = (const float*)d_in[10];
  const float* b_q2  = (const float*)d_in[11];
  float* out = (float*)d_out;

  float* ws  = (float*)d_ws;
  float* qt2 = ws;                  // 32*100
  float* qt1 = qt2 + Bv * Dv;
  float* qw2 = qt1 + Bv * Dv;
  float* qw1 = qw2 + Bv * Dv;

  proj_tanh_wmma<<<28, 32, 0, stream>>>(q, W_q2, b_q2, W_q1, b_q1, qt2, qt1);
  qw_wmma<<<28, 32, 0, stream>>>(qt2, qt1, W_kv2, W_kv1, qw2, qw1);
  wordgraph_attn_main<<<Bv * N1v, 256, 0, stream>>>(k_1, v_1, k_2, v_2,
                                                    qw2, qw1, out);
}